// ASIS_38792144617768
// MI455X (gfx1250) — compile-verified
//
#include <hip/hip_runtime.h>
#include <stdint.h>

typedef float v2f __attribute__((ext_vector_type(2)));
typedef float v8f __attribute__((ext_vector_type(8)));

#define T_PTS   16384
#define C_CH    256
#define CI_CH   8
#define E_DIM   5
#define NC_CLS  13
#define N_PER   8192
#define KNN     20
#define TH2     0.25f
#define BN_EPS  1e-5f

#define AS_STRIDE 36     // A row stride (floats): 144B, 16B-aligned rows, conflict-free frags
#define BS_STRIDE 160    // B k-pair row stride (floats): half-waves hit disjoint bank halves

// ---------------------------------------------------------------------------
// CDNA5 async global->LDS helpers (tracked by ASYNCcnt, no VGPR round-trip)
// ---------------------------------------------------------------------------
__device__ __forceinline__ void async_lds_b128(uint32_t lds_off, const void* gaddr) {
    asm volatile("global_load_async_to_lds_b128 %0, %1, off"
                 :: "v"(lds_off), "v"(gaddr) : "memory");
}
__device__ __forceinline__ void async_lds_b32(uint32_t lds_off, const void* gaddr) {
    asm volatile("global_load_async_to_lds_b32 %0, %1, off"
                 :: "v"(lds_off), "v"(gaddr) : "memory");
}
__device__ __forceinline__ void wait_async0() {
    asm volatile("s_wait_asynccnt 0x0" ::: "memory");
}

// ---------------------------------------------------------------------------
// Kernel 1: f_sins tile GEMM (WMMA f32 16x16x4), async double-buffered LDS
// staging, BN/ReLU epilogue for both branches, per-tile W_emb projection.
// Block: 256 threads (8 waves). Tile: 128 rows x 64 cols. K=256 in 8 chunks.
// ---------------------------------------------------------------------------
__global__ __launch_bounds__(256) void asis_gemm_emb(
    const float* __restrict__ f_sem, const float* __restrict__ f_ins,
    const float* __restrict__ W_sem, const float* __restrict__ b_sem,
    const float* __restrict__ g_sem, const float* __restrict__ beta_sem,
    const float* __restrict__ m_sem, const float* __restrict__ v_sem,
    const float* __restrict__ W_ins, const float* __restrict__ b_ins,
    const float* __restrict__ g_ins, const float* __restrict__ beta_ins,
    const float* __restrict__ m_ins, const float* __restrict__ v_ins,
    const float* __restrict__ W_emb, float* __restrict__ e_part)
{
    // double-buffered stage overlaid with the epilogue f_sins tile
    __shared__ __align__(16) union SMem {
        struct {
            float As[2][128 * AS_STRIDE];   // 2 x 18432 B
            float Bs[2][16 * BS_STRIDE];    // 2 x 10240 B
        } st;
        float Fs[128 * 65];                 // 33280 B (used after k-loop only)
    } smem;
    __shared__ float FinsS[128 * 8];

    const int tid  = threadIdx.x;
    const int lane = tid & 31;
    const int wave = tid >> 5;
    const int l16  = lane & 15;
    const int hs   = lane >> 4;          // half-wave select (k pair / M+8)
    const int R0   = blockIdx.x * 128;
    const int C0   = blockIdx.y * 64;
    const int r0   = wave * 16;          // wave's 16-row strip

    // stage f_ins rows (contiguous 128*8 floats); visible after first barrier
    #pragma unroll
    for (int i = 0; i < 4; ++i) {
        int idx = tid + i * 256;
        FinsS[idx] = f_ins[R0 * CI_CH + idx];
    }

    // async-issue one K-chunk into buffer b
    auto issue_chunk = [&](int kc, int b) {
        const int Kb = kc * 32;
        // A tile 128x32: b128 per lane (16B aligned both sides)
        #pragma unroll
        for (int i = 0; i < 4; ++i) {
            int idx = tid + i * 256;
            int r  = idx >> 3;
            int kq = (idx & 7) << 2;
            uint32_t lo = (uint32_t)(uintptr_t)&smem.st.As[b][r * AS_STRIDE + kq];
            async_lds_b128(lo, f_sem + (size_t)(R0 + r) * C_CH + Kb + kq);
        }
        // B tile 32x64: b32 per element into k-pair-interleaved layout
        #pragma unroll
        for (int i = 0; i < 8; ++i) {
            int idx = tid + i * 256;
            int kk = idx >> 6, c = idx & 63;
            uint32_t lo = (uint32_t)(uintptr_t)
                &smem.st.Bs[b][(kk >> 1) * BS_STRIDE + c * 2 + (kk & 1)];
            async_lds_b32(lo, W_sem + (size_t)(Kb + kk) * C_CH + C0 + c);
        }
    };

    const v8f vzero = {0.f, 0.f, 0.f, 0.f, 0.f, 0.f, 0.f, 0.f};
    v8f acc[4];
    #pragma unroll
    for (int i = 0; i < 4; ++i) acc[i] = vzero;

    issue_chunk(0, 0);
    for (int kc = 0; kc < 8; ++kc) {
        const int cur = kc & 1;
        wait_async0();          // this wave's chunk-kc transfers landed
        __syncthreads();        // everyone's landed; prev buffer fully consumed
        if (kc + 1 < 8) issue_chunk(kc + 1, cur ^ 1);   // overlap with compute

        const float* Asb = smem.st.As[cur];
        const float* Bsb = smem.st.Bs[cur];
        #pragma unroll
        for (int kb = 0; kb < 32; kb += 4) {
            // A frag: lanes 0-15 k{kb,kb+1}, lanes 16-31 k{kb+2,kb+3} (adjacent b64)
            v2f a = *(const v2f*)&Asb[(r0 + l16) * AS_STRIDE + kb + 2 * hs];
            #pragma unroll
            for (int i = 0; i < 4; ++i) {
                // B frag: adjacent pair in interleaved layout (single b64)
                v2f b = *(const v2f*)
                    &Bsb[((kb >> 1) + hs) * BS_STRIDE + (i * 16 + l16) * 2];
                acc[i] = __builtin_amdgcn_wmma_f32_16x16x4_f32(
                    false, a, false, b, (short)0, acc[i], false, false);
            }
        }
        __syncthreads();        // all waves done with buffer `cur`
    }

    // Epilogue: BN+ReLU (sem) + inline f_ins branch -> f_sins tile in LDS.
    // C/D layout: VGPR v -> row r0+v (lanes 0-15) or r0+v+8 (lanes 16-31).
    #pragma unroll
    for (int i = 0; i < 4; ++i) {
        const int c   = C0 + i * 16 + l16;
        const float bs_ = b_sem[c];
        const float rs  = rsqrtf(v_sem[c] + BN_EPS);
        const float gs  = g_sem[c], bts = beta_sem[c], ms = m_sem[c];
        const float ri  = rsqrtf(v_ins[c] + BN_EPS);
        const float gi  = g_ins[c], bti = beta_ins[c], mi = m_ins[c];
        const float bii = b_ins[c];
        float wi[8];
        #pragma unroll
        for (int k = 0; k < 8; ++k) wi[k] = W_ins[k * C_CH + c];

        #pragma unroll
        for (int vv = 0; vv < 8; ++vv) {
            const int rl = r0 + vv + 8 * hs;
            float x  = acc[i][vv] + bs_;
            float fs = fmaxf(gs * (x - ms) * rs + bts, 0.0f);
            const float* fr = &FinsS[rl * 8];
            float y = bii;
            #pragma unroll
            for (int k = 0; k < 8; ++k) y += fr[k] * wi[k];
            float fh = fmaxf(gi * (y - mi) * ri + bti, 0.0f);
            smem.Fs[rl * 65 + i * 16 + l16] = fs + fh;
        }
    }
    __syncthreads();

    // Project the 64-col tile onto W_emb: e_part[colBlk][row][0..4]
    {
        const int row  = tid >> 1;
        const int half = tid & 1;
        float s0 = 0.f, s1 = 0.f, s2 = 0.f, s3 = 0.f, s4 = 0.f;
        const float* frow = &smem.Fs[row * 65 + half * 32];
        const float* wb   = &W_emb[(C0 + half * 32) * E_DIM];
        #pragma unroll 4
        for (int c = 0; c < 32; ++c) {
            float fv = frow[c];
            s0 += fv * wb[c * E_DIM + 0];
            s1 += fv * wb[c * E_DIM + 1];
            s2 += fv * wb[c * E_DIM + 2];
            s3 += fv * wb[c * E_DIM + 3];
            s4 += fv * wb[c * E_DIM + 4];
        }
        s0 += __shfl_xor(s0, 1); s1 += __shfl_xor(s1, 1);
        s2 += __shfl_xor(s2, 1); s3 += __shfl_xor(s3, 1);
        s4 += __shfl_xor(s4, 1);
        if (half == 0) {
            float* ep = &e_part[((size_t)blockIdx.y * T_PTS + R0 + row) * E_DIM];
            ep[0] = s0; ep[1] = s1; ep[2] = s2; ep[3] = s3; ep[4] = s4;
        }
    }
}

// ---------------------------------------------------------------------------
// Kernel 2: deterministic reduction of the 4 column-block partials,
// add b_emb, emit e_ins to d_out and row squared-norms to workspace.
// ---------------------------------------------------------------------------
__global__ __launch_bounds__(256) void asis_emb_finalize(
    const float* __restrict__ e_part, const float* __restrict__ b_emb,
    float* __restrict__ e_out, float* __restrict__ sq)
{
    const int t = blockIdx.x * blockDim.x + threadIdx.x;
    if (t >= T_PTS) return;
    float s = 0.f;
    #pragma unroll
    for (int d = 0; d < E_DIM; ++d) {
        float v = b_emb[d];
        #pragma unroll
        for (int cb = 0; cb < 4; ++cb)
            v += e_part[((size_t)cb * T_PTS + t) * E_DIM + d];
        e_out[t * E_DIM + d] = v;
        s += v * v;
    }
    sq[t] = s;
}

// ---------------------------------------------------------------------------
// Kernel 3: one wave per point. Threshold-gated K-NN over the 8192-point
// cloud, exact top-20 selection (rare slow path), max-gather of f_sem over
// the neighbor set, fused 13-way classifier with butterfly reduction.
// ---------------------------------------------------------------------------
__global__ __launch_bounds__(256) void asis_fusion_cls(
    const float* __restrict__ f_sem, const float* __restrict__ e,
    const float* __restrict__ sq, const float* __restrict__ W_cls,
    const float* __restrict__ b_cls, float* __restrict__ p_out)
{
    __shared__ float sD[8 * 640];
    __shared__ int   sI[8 * 640];
    __shared__ int   sCnt[8];

    const int tid  = threadIdx.x;
    const int lane = tid & 31;
    const int wave = tid >> 5;
    const int i    = blockIdx.x * 8 + wave;
    const int base = (i < N_PER) ? 0 : N_PER;

    const float e0 = e[i * E_DIM + 0], e1 = e[i * E_DIM + 1];
    const float e2 = e[i * E_DIM + 2], e3 = e[i * E_DIM + 3];
    const float e4 = e[i * E_DIM + 4];
    const float sqi = sq[i];

    float dist[KNN];
    int   idx[KNN];
    int cnt = 0, hits = 0;

    for (int jj = 0; jj < N_PER / 32; ++jj) {
        const int j = base + jj * 32 + lane;
        const float* ej = &e[j * E_DIM];
        float dot = e0 * ej[0] + e1 * ej[1] + e2 * ej[2] + e3 * ej[3] + e4 * ej[4];
        float d2  = fmaxf(sqi + sq[j] - 2.0f * dot, 0.0f);
        if (d2 <= TH2) {
            ++hits;
            if (cnt < KNN) {
                dist[cnt] = d2; idx[cnt] = j; ++cnt;
            } else {
                float w = dist[0]; int wp = 0;
                #pragma unroll
                for (int t = 1; t < KNN; ++t)
                    if (dist[t] > w) { w = dist[t]; wp = t; }
                if (d2 < w) { dist[wp] = d2; idx[wp] = j; }
            }
        }
    }

    // cross-lane totals + compaction offsets
    int totalHits = hits;
    #pragma unroll
    for (int s = 16; s; s >>= 1) totalHits += __shfl_xor(totalHits, s);
    int pre = cnt;
    #pragma unroll
    for (int s = 1; s < 32; s <<= 1) {
        int t2 = __shfl_up(pre, s);
        if (lane >= s) pre += t2;
    }
    const int excl     = pre - cnt;
    const int totalRet = __shfl(pre, 31);

    float* Dw = &sD[wave * 640];
    int*   Iw = &sI[wave * 640];
    for (int t = 0; t < cnt; ++t) { Dw[excl + t] = dist[t]; Iw[excl + t] = idx[t]; }
    __syncthreads();

    if (lane == 0) {
        if (totalHits > KNN) {
            // exact 20-smallest among retained candidates (rare path)
            const int m = totalRet;
            for (int t = 0; t < KNN; ++t) {
                int best = t; float bd = Dw[t];
                for (int u = t + 1; u < m; ++u)
                    if (Dw[u] < bd) { bd = Dw[u]; best = u; }
                float td = Dw[t]; Dw[t] = Dw[best]; Dw[best] = td;
                int   ti = Iw[t]; Iw[t] = Iw[best]; Iw[best] = ti;
            }
            sCnt[wave] = KNN;
        } else {
            sCnt[wave] = totalRet;   // all hits retained, no truncation
        }
    }
    __syncthreads();

    // max-gather of f_sem rows: lane owns 8 channels (two float4s)
    const int count = sCnt[wave];
    const int c0    = lane * 8;
    float mx[8];
    #pragma unroll
    for (int k = 0; k < 8; ++k) mx[k] = -3.4e38f;
    for (int t = 0; t < count; ++t) {
        const int j = Iw[t];
        const float4* fj = (const float4*)&f_sem[(size_t)j * C_CH + c0];
        float4 u0 = fj[0], u1 = fj[1];
        mx[0] = fmaxf(mx[0], u0.x); mx[1] = fmaxf(mx[1], u0.y);
        mx[2] = fmaxf(mx[2], u0.z); mx[3] = fmaxf(mx[3], u0.w);
        mx[4] = fmaxf(mx[4], u1.x); mx[5] = fmaxf(mx[5], u1.y);
        mx[6] = fmaxf(mx[6], u1.z); mx[7] = fmaxf(mx[7], u1.w);
    }

    // fused classifier: partial over this lane's 8 channels, butterfly-reduce
    float p[NC_CLS];
    #pragma unroll
    for (int n = 0; n < NC_CLS; ++n) p[n] = 0.f;
    #pragma unroll
    for (int k = 0; k < 8; ++k) {
        const float* wr = &W_cls[(size_t)(c0 + k) * NC_CLS];
        const float fv  = mx[k];
        #pragma unroll
        for (int n = 0; n < NC_CLS; ++n) p[n] += fv * wr[n];
    }
    #pragma unroll
    for (int n = 0; n < NC_CLS; ++n) {
        #pragma unroll
        for (int s = 16; s; s >>= 1) p[n] += __shfl_xor(p[n], s);
    }
    if (lane < NC_CLS) p_out[(size_t)i * NC_CLS + lane] = p[lane] + b_cls[lane];
}

// ---------------------------------------------------------------------------
extern "C" void kernel_launch(void* const* d_in, const int* in_sizes, int n_in,
                              void* d_out, int out_size, void* d_ws, size_t ws_size,
                              hipStream_t stream) {
    const float* f_sem    = (const float*)d_in[0];
    const float* f_ins    = (const float*)d_in[1];
    // d_in[2] = batch (int64): layout is known (2 equal sorted clouds), unused.
    const float* W_sem    = (const float*)d_in[3];
    const float* b_sem    = (const float*)d_in[4];
    const float* g_sem    = (const float*)d_in[5];
    const float* beta_sem = (const float*)d_in[6];
    const float* m_sem    = (const float*)d_in[7];
    const float* v_sem    = (const float*)d_in[8];
    const float* W_ins    = (const float*)d_in[9];
    const float* b_ins    = (const float*)d_in[10];
    const float* g_ins    = (const float*)d_in[11];
    const float* beta_ins = (const float*)d_in[12];
    const float* m_ins    = (const float*)d_in[13];
    const float* v_ins    = (const float*)d_in[14];
    const float* W_emb    = (const float*)d_in[15];
    const float* b_emb    = (const float*)d_in[16];
    const float* W_cls    = (const float*)d_in[17];
    const float* b_cls    = (const float*)d_in[18];

    float* p_out  = (float*)d_out;             // [16384 x 13]
    float* e_out  = p_out + T_PTS * NC_CLS;    // [16384 x 5]
    float* e_part = (float*)d_ws;              // [4 x 16384 x 5]
    float* sq     = e_part + 4 * T_PTS * E_DIM;// [16384]

    asis_gemm_emb<<<dim3(T_PTS / 128, C_CH / 64), dim3(256), 0, stream>>>(
        f_sem, f_ins, W_sem, b_sem, g_sem, beta_sem, m_sem, v_sem,
        W_ins, b_ins, g_ins, beta_ins, m_ins, v_ins, W_emb, e_part);

    asis_emb_finalize<<<dim3((T_PTS + 255) / 256), dim3(256), 0, stream>>>(
        e_part, b_emb, e_out, sq);

    asis_fusion_cls<<<dim3(T_PTS / 8), dim3(256), 0, stream>>>(
        f_sem, e_out, sq, W_cls, b_cls, p_out);
}